// TabularEmbeddingDecoder_88974542504535
// MI455X (gfx1250) — compile-verified
//
#include <hip/hip_runtime.h>
#include <hip/hip_bf16.h>
#include <math.h>

typedef __attribute__((ext_vector_type(2))) float v2f;
typedef __attribute__((ext_vector_type(8))) float v8f;

constexpr int kNRows     = 8192;        // B*T
constexpr int kF         = 188;
constexpr int kH         = 16;
constexpr int kRowStride = kF * kH;     // 3008 floats per row of x
constexpr int kNCont     = 177;

// Fast device transcendentals: native TRANS32 ops instead of branchy libm.
__device__ __forceinline__ float fast_rcp(float a) {
    return __builtin_amdgcn_rcpf(a);            // v_rcp_f32
}
__device__ __forceinline__ float fast_sigmoid(float a) {
    return fast_rcp(1.0f + __expf(-a));         // v_exp_f32 + v_rcp_f32
}
__device__ __forceinline__ float fast_tanh(float a) {
#if __has_builtin(__builtin_amdgcn_tanhf)
    return __builtin_amdgcn_tanhf(a);           // v_tanh_f32
#elif __has_builtin(__builtin_amdgcn_tanh_f32)
    return __builtin_amdgcn_tanh_f32(a);
#else
    // tanh(|a|) = (1 - e^{-2|a|}) / (1 + e^{-2|a|}), sign-restored
    const float e2 = __expf(-2.0f * __builtin_fabsf(a));
    const float t  = (1.0f - e2) * fast_rcp(1.0f + e2);
    return __builtin_copysignf(t, a);
#endif
}

// One block = 4 waves = one 16-row tile. Each wave handles features
// f = waveId, waveId+4, ... (47 features per wave; 188 = 4*47, so all
// waves run the same trip count -> EXEC is all-ones at every WMMA).
__global__ __launch_bounds__(128)
void tab_decoder_wmma(const float* __restrict__ x,
                      const float* __restrict__ W2a, const float* __restrict__ b2a,
                      const float* __restrict__ W3,  const float* __restrict__ b3,
                      const float* __restrict__ W4,  const float* __restrict__ b4,
                      const float* __restrict__ W6,  const float* __restrict__ b6,
                      const float* __restrict__ W7,  const float* __restrict__ b7,
                      const float* __restrict__ Wc,  const float* __restrict__ bc,
                      const float* __restrict__ eps,
                      float* __restrict__ out)
{
    const int tile = blockIdx.x;            // 512 tiles of 16 rows
    const int wave = threadIdx.x >> 5;      // 0..3
    const int lane = threadIdx.x & 31;
    const int m    = lane & 15;             // A-row / C-col index within half
    const int kh   = lane >> 4;             // 0 = lanes 0-15, 1 = lanes 16-31
    const int n    = m;                     // B/C/D column index for this lane

    const int    arow = tile * 16 + m;      // row this lane supplies to A
    const float* xb   = x + (size_t)arow * kRowStride;

    for (int f = wave; f < kF; f += 4) {
        // ---- select per-feature weight / bias / out-dim (wave-uniform) ----
        const float* W; const float* bias; int od;
        if      (f < 4)  { W = W2a + f       * kH * 2; bias = b2a + f       * 2; od = 2; }
        else if (f < 6)  { W = W3  + (f - 4) * kH * 3; bias = b3  + (f - 4) * 3; od = 3; }
        else if (f < 7)  { W = W4;                     bias = b4;                od = 4; }
        else if (f < 8)  { W = W6;                     bias = b6;                od = 6; }
        else if (f < 11) { W = W7  + (f - 8) * kH * 7; bias = b7  + (f - 8) * 7; od = 7; }
        else             { W = Wc  + (f - 11)* kH * 2; bias = bc  + (f - 11)* 2; od = 2; }

        const float* xf = xb + f * kH;
        // prefetch this lane's A data for the next feature this wave handles
        __builtin_prefetch(xf + 4 * kH, 0, 1);

        const bool valid = (n < od);
        const int  nc    = valid ? n : 0;       // clamped, always in-bounds column

        // ---- unconditional loads (clamped addr) + cndmask select: no exec
        //      divergence, so no s_and_saveexec blocks around the loads ----
        const float braw = bias[nc];
        const float bn   = valid ? braw : 0.0f;

        // A tile: this lane's K pairs for all 4 WMMA steps (4x global_load_b64)
        v2f a0 = *(const v2f*)(xf + 0  + 2 * kh);
        v2f a1 = *(const v2f*)(xf + 4  + 2 * kh);
        v2f a2 = *(const v2f*)(xf + 8  + 2 * kh);
        v2f a3 = *(const v2f*)(xf + 12 + 2 * kh);

        // B tile: 8 scalars, loaded unconditionally from clamped column
        const float* Wl = W + 2 * kh * od + nc; // row (2*kh), column nc
        float w0 = Wl[0 * od], w1 = Wl[1 * od];
        float w2 = Wl[4 * od], w3 = Wl[5 * od];
        float w4 = Wl[8 * od], w5 = Wl[9 * od];
        float w6 = Wl[12 * od], w7 = Wl[13 * od];

        v2f b0, b1, b2, b3;
        b0.x = valid ? w0 : 0.0f;  b0.y = valid ? w1 : 0.0f;
        b1.x = valid ? w2 : 0.0f;  b1.y = valid ? w3 : 0.0f;
        b2.x = valid ? w4 : 0.0f;  b2.y = valid ? w5 : 0.0f;
        b3.x = valid ? w6 : 0.0f;  b3.y = valid ? w7 : 0.0f;

        // ---- D = A(16x16) * B(16xod padded to 16): 4 chained K=4 WMMAs ----
        v8f acc = {};
        acc = __builtin_amdgcn_wmma_f32_16x16x4_f32(false, a0, false, b0,
                                                    (short)0, acc, false, false);
        acc = __builtin_amdgcn_wmma_f32_16x16x4_f32(false, a1, false, b1,
                                                    (short)0, acc, false, false);
        acc = __builtin_amdgcn_wmma_f32_16x16x4_f32(false, a2, false, b2,
                                                    (short)0, acc, false, false);
        acc = __builtin_amdgcn_wmma_f32_16x16x4_f32(false, a3, false, b3,
                                                    (short)0, acc, false, false);

        // C/D layout: acc[r] holds row (tile*16 + r + 8*kh), column n = lane%16.
        if (f < 11) {
            // ---- categorical: argmax over columns 0..od-1 (od <= 7) ----
#pragma unroll
            for (int r = 0; r < 8; ++r) {
                float v   = valid ? (acc[r] + bn) : -3.0e38f;
                int   idx = n;
#pragma unroll
                for (int mask = 1; mask <= 4; mask <<= 1) {
                    float ov = __shfl_xor(v,   mask, 32);
                    int   oi = __shfl_xor(idx, mask, 32);
                    // jnp.argmax: max value, lowest index on ties
                    if (ov > v || (ov == v && oi < idx)) { v = ov; idx = oi; }
                }
                if (n == 0) {
                    const int orow = tile * 16 + r + 8 * kh;
                    out[(size_t)orow * kF + f] = (float)idx;
                }
            }
        } else {
            // ---- continuous: mean=tanh(col0), std=sigmoid(col1), samp=mean+std*eps ----
            const int fc = f - 11;
#pragma unroll
            for (int r = 0; r < 8; ++r) {
                float t  = acc[r] + bn;                 // n==0: mean logit, n==1: std logit
                float ts = __shfl_xor(t, 1, 32);        // lane0 gets lane1's std logit
                if (n == 0) {
                    const int orow = tile * 16 + r + 8 * kh;
                    const float mean = fast_tanh(t);
                    const float sd   = fast_sigmoid(ts);
                    const float e    = eps[(size_t)orow * kNCont + fc];
                    out[(size_t)orow * kF + f] = mean + sd * e;
                }
            }
        }
    }
}

extern "C" void kernel_launch(void* const* d_in, const int* in_sizes, int n_in,
                              void* d_out, int out_size, void* d_ws, size_t ws_size,
                              hipStream_t stream) {
    const float* x   = (const float*)d_in[0];
    const float* W2a = (const float*)d_in[1];
    const float* b2a = (const float*)d_in[2];
    const float* W3  = (const float*)d_in[3];
    const float* b3  = (const float*)d_in[4];
    const float* W4  = (const float*)d_in[5];
    const float* b4  = (const float*)d_in[6];
    const float* W6  = (const float*)d_in[7];
    const float* b6  = (const float*)d_in[8];
    const float* W7  = (const float*)d_in[9];
    const float* b7  = (const float*)d_in[10];
    const float* Wc  = (const float*)d_in[11];
    const float* bc  = (const float*)d_in[12];
    const float* eps = (const float*)d_in[13];
    float* out = (float*)d_out;

    // 8192 rows / 16 rows per tile = 512 blocks; 4 waves per block.
    tab_decoder_wmma<<<dim3(kNRows / 16), dim3(128), 0, stream>>>(
        x, W2a, b2a, W3, b3, W4, b4, W6, b6, W7, b7, Wc, bc, eps, out);
}